// KNNGraphSAGE_87952340287519
// MI455X (gfx1250) — compile-verified
//
#include <hip/hip_runtime.h>
#include <hip/hip_bf16.h>

typedef __attribute__((ext_vector_type(2))) float v2f;
typedef __attribute__((ext_vector_type(8))) float v8f;

#define N_NODES 100000
#define N_EDGES 3200000
#define DIM     128
#define EPSBN   1e-5f

// ---------------- scatter-add: mean-aggregation numerator + degree ----------
// one wave (32 lanes) per edge; each lane moves a float4 (32*4 = 128 feats)
__global__ void scatter_add_kernel(const float* __restrict__ X,
                                   const int* __restrict__ src,
                                   const int* __restrict__ tgt,
                                   float* __restrict__ ssum,
                                   float* __restrict__ deg) {
  long long tid = (long long)blockIdx.x * blockDim.x + threadIdx.x;
  int e    = (int)(tid >> 5);
  int lane = (int)(tid & 31);
  if (e >= N_EDGES) return;
  int s = src[e];
  int t = tgt[e];
  float4 v = *((const float4*)(X + (size_t)s * DIM) + lane);
  float* dst = ssum + (size_t)t * DIM + lane * 4;
  atomicAdd(dst + 0, v.x);
  atomicAdd(dst + 1, v.y);
  atomicAdd(dst + 2, v.z);
  atomicAdd(dst + 3, v.w);
  if (lane == 0) atomicAdd(deg + t, 1.0f);
}

// ---------------- mean = ssum / max(deg,1) ---------------------------------
__global__ void mean_div_kernel(float* __restrict__ ssum,
                                const float* __restrict__ deg) {
  long long i = (long long)blockIdx.x * blockDim.x + threadIdx.x;
  if (i >= (long long)N_NODES * DIM) return;
  int row = (int)(i >> 7);
  ssum[i] = ssum[i] / fmaxf(deg[row], 1.0f);
}

// ---------------- fused SAGE GEMM:  Out = A@Wn (+ X@Wr) + bias  (WMMA f32) --
// one wave per 16x16 output tile; V_WMMA_F32_16X16X4_F32, K stepped by 4.
// A 16x4 frag: M = lane&15, K = k0 + 2*(lane>=16) (+1 for 2nd VGPR)
// B 4x16 frag: N = lane&15, same K split
// C/D:         row = v + 8*(lane>=16), col = lane&15
template <bool HAS_ROOT, bool RELU>
__global__ void sage_gemm_wmma(const float* __restrict__ A,    // [M,K] aggregated
                               const float* __restrict__ X,    // [M,K] root (or null)
                               const float* __restrict__ Wn,   // [K,N]
                               const float* __restrict__ Wr,   // [K,N] (or null)
                               const float* __restrict__ bias, // [N]
                               float* __restrict__ Out,        // [M,N]
                               int M, int K, int N) {
  int wave = threadIdx.x >> 5;
  int lane = threadIdx.x & 31;
  int tilesN = N >> 4;
  int totalTiles = (M >> 4) * tilesN;
  int tile = blockIdx.x * (blockDim.x >> 5) + wave;
  if (tile >= totalTiles) return;            // wave-uniform: EXEC stays all-1s
  int tm = tile / tilesN;
  int tn = tile - tm * tilesN;
  int hi = lane >> 4;                        // 0: K=k0,k0+1   1: K=k0+2,k0+3
  int lr = lane & 15;

  const float* Arow = A + (size_t)(tm * 16 + lr) * K;
  const float* Xrow = HAS_ROOT ? (X + (size_t)(tm * 16 + lr) * K) : nullptr;
  const float* Bn   = Wn + tn * 16 + lr;
  const float* Br   = HAS_ROOT ? (Wr + tn * 16 + lr) : nullptr;

  v8f acc = {};
  for (int k0 = 0; k0 < K; k0 += 4) {
    int k = k0 + 2 * hi;
    v2f a; a.x = Arow[k];             a.y = Arow[k + 1];
    v2f b; b.x = Bn[(size_t)k * N];   b.y = Bn[(size_t)(k + 1) * N];
    acc = __builtin_amdgcn_wmma_f32_16x16x4_f32(false, a, false, b,
                                                (short)0, acc, false, false);
    if (HAS_ROOT) {
      v2f ar; ar.x = Xrow[k];            ar.y = Xrow[k + 1];
      v2f br; br.x = Br[(size_t)k * N];  br.y = Br[(size_t)(k + 1) * N];
      acc = __builtin_amdgcn_wmma_f32_16x16x4_f32(false, ar, false, br,
                                                  (short)0, acc, false, false);
    }
  }
  float bv = bias[tn * 16 + lr];
#pragma unroll
  for (int v = 0; v < 8; ++v) {
    float val = acc[v] + bv;
    if (RELU) val = fmaxf(val, 0.0f);
    Out[(size_t)(tm * 16 + v + 8 * hi) * N + tn * 16 + lr] = val;
  }
}

// ---------------- batchnorm: per-feature partial sums ----------------------
__global__ void bn_stats_kernel(const float* __restrict__ H,
                                float* __restrict__ sums,
                                float* __restrict__ sq,
                                int rowsPerBlock) {
  int f  = threadIdx.x;                    // 0..127
  int r0 = blockIdx.x * rowsPerBlock;
  int r1 = min(r0 + rowsPerBlock, N_NODES);
  float s = 0.f, s2 = 0.f;
  for (int r = r0; r < r1; ++r) {
    float v = H[(size_t)r * DIM + f];
    s += v;
    s2 += v * v;
  }
  atomicAdd(&sums[f], s);
  atomicAdd(&sq[f], s2);
}

__global__ void bn_finalize_kernel(const float* __restrict__ sums,
                                   const float* __restrict__ sq,
                                   const float* __restrict__ g,
                                   const float* __restrict__ be,
                                   float* __restrict__ scale,
                                   float* __restrict__ shift) {
  int f = threadIdx.x;
  float invN = 1.0f / (float)N_NODES;
  float mu  = sums[f] * invN;
  float var = sq[f] * invN - mu * mu;       // biased variance (torch BN training)
  float rs  = rsqrtf(var + EPSBN);
  float sc  = g[f] * rs;
  scale[f] = sc;
  shift[f] = be[f] - mu * sc;
}

__global__ void bn_apply_relu_kernel(float* __restrict__ H,
                                     const float* __restrict__ scale,
                                     const float* __restrict__ shift) {
  long long i = (long long)blockIdx.x * blockDim.x + threadIdx.x;
  if (i >= (long long)N_NODES * DIM) return;
  int f = (int)(i & (DIM - 1));
  H[i] = fmaxf(H[i] * scale[f] + shift[f], 0.0f);
}

// ---------------- final 64 -> 1 projection ---------------------------------
__global__ void final_dot_kernel(const float* __restrict__ H2,  // [N,64], relu'd
                                 const float* __restrict__ Wc2, // [64]
                                 const float* __restrict__ bc2, // [1]
                                 float* __restrict__ out) {
  int i = blockIdx.x * blockDim.x + threadIdx.x;
  if (i >= N_NODES) return;
  const float* h = H2 + (size_t)i * 64;
  float acc = bc2[0];
#pragma unroll
  for (int j = 0; j < 64; ++j) acc += h[j] * Wc2[j];
  out[i] = acc;
}

extern "C" void kernel_launch(void* const* d_in, const int* in_sizes, int n_in,
                              void* d_out, int out_size, void* d_ws, size_t ws_size,
                              hipStream_t stream) {
  const float* x    = (const float*)d_in[0];
  const int*   ei   = (const int*)d_in[1];           // [2, E] int32
  const float* Wn0  = (const float*)d_in[2];
  const float* b0   = (const float*)d_in[3];
  const float* Wr0  = (const float*)d_in[4];
  const float* g0   = (const float*)d_in[5];
  const float* be0  = (const float*)d_in[6];
  const float* Wn1  = (const float*)d_in[7];
  const float* b1   = (const float*)d_in[8];
  const float* Wr1  = (const float*)d_in[9];
  const float* g1   = (const float*)d_in[10];
  const float* be1  = (const float*)d_in[11];
  const float* Wc1  = (const float*)d_in[12];
  const float* bc1  = (const float*)d_in[13];
  const float* Wc2  = (const float*)d_in[14];
  const float* bc2  = (const float*)d_in[15];
  const int* src = ei;
  const int* tgt = ei + N_EDGES;

  // workspace layout (floats): A | B | C | deg | sums(128) sq(128) scale(128) shift(128)
  float* bufA  = (float*)d_ws;                               // aggregation / classifier hidden
  float* bufB  = bufA + (size_t)N_NODES * DIM;               // layer-0 activations
  float* bufC  = bufB + (size_t)N_NODES * DIM;               // layer-1 activations
  float* deg   = bufC + (size_t)N_NODES * DIM;
  float* sums  = deg + N_NODES;
  float* sq    = sums + DIM;
  float* scale = sq + DIM;
  float* shift = scale + DIM;

  const int EW_BLOCKS   = (N_NODES * DIM) / 256;             // 50000
  const int SC_BLOCKS   = (N_EDGES * 32) / 256;              // 400000
  const int GEMM_TILES  = (N_NODES / 16) * (DIM / 16);       // 50000 (div by 8)
  const int BN_ROWS     = 250;
  const int BN_BLOCKS   = (N_NODES + BN_ROWS - 1) / BN_ROWS; // 400

  auto layer = [&](const float* xin, const float* Wn, const float* b,
                   const float* Wr, const float* g, const float* be, float* hout) {
    hipMemsetAsync(bufA, 0, (size_t)N_NODES * DIM * sizeof(float), stream);
    hipMemsetAsync(deg, 0, (size_t)(N_NODES + 2 * DIM) * sizeof(float), stream); // deg+sums+sq
    scatter_add_kernel<<<SC_BLOCKS, 256, 0, stream>>>(xin, src, tgt, bufA, deg);
    mean_div_kernel<<<EW_BLOCKS, 256, 0, stream>>>(bufA, deg);
    sage_gemm_wmma<true, false><<<GEMM_TILES / 8, 256, 0, stream>>>(
        bufA, xin, Wn, Wr, b, hout, N_NODES, DIM, DIM);
    bn_stats_kernel<<<BN_BLOCKS, DIM, 0, stream>>>(hout, sums, sq, BN_ROWS);
    bn_finalize_kernel<<<1, DIM, 0, stream>>>(sums, sq, g, be, scale, shift);
    bn_apply_relu_kernel<<<EW_BLOCKS, 256, 0, stream>>>(hout, scale, shift);
  };

  layer(x,    Wn0, b0, Wr0, g0, be0, bufB);   // layer 0
  layer(bufB, Wn1, b1, Wr1, g1, be1, bufC);   // layer 1

  // classifier: relu(h @ Wc1 + bc1) -> [N,64] into bufA (free now)
  const int CLS_TILES = (N_NODES / 16) * (64 / 16);          // 25000 (div by 8)
  sage_gemm_wmma<false, true><<<CLS_TILES / 8, 256, 0, stream>>>(
      bufC, nullptr, Wc1, nullptr, bc1, bufA, N_NODES, DIM, 64);

  final_dot_kernel<<<(N_NODES + 255) / 256, 256, 0, stream>>>(
      bufA, Wc2, bc2, (float*)d_out);
}